// ImageBlendSynth_4002909520631
// MI455X (gfx1250) — compile-verified
//
#include <hip/hip_runtime.h>

typedef float v2f __attribute__((ext_vector_type(2)));
typedef float v8f __attribute__((ext_vector_type(8)));

#define IMG_H 256
#define IMG_W 1024
#define NB    256
#define WSPLIT 4                       // w-tile range split per b
#define WT_PER_BLK (IMG_W / 16 / WSPLIT)  // 16 w-tiles per block

// ---------------------------------------------------------------------------
// Kernel 0: init ws for atomic min/max (uint-ordered; target is in [0,1))
// ---------------------------------------------------------------------------
__global__ void init_ws_kernel(unsigned int* __restrict__ wsu) {
    wsu[0] = 0xFFFFFFFFu;  // running min (as uint bits)
    wsu[1] = 0u;           // running max (as uint bits)
}

// ---------------------------------------------------------------------------
// Kernel 1: global min/max of target image -> ws[0]=min, ws[1]=max
// 64 blocks, per-block LDS reduce, leader does atomicMin/Max on float bits
// (valid ordering because target values are non-negative).
// ---------------------------------------------------------------------------
__global__ __launch_bounds__(256) void minmax_kernel(const float* __restrict__ t,
                                                     unsigned int* __restrict__ wsu) {
    __shared__ float smin[256];
    __shared__ float smax[256];
    const int tid = threadIdx.x;
    const int gid = blockIdx.x * 256 + tid;
    const float4* t4 = (const float4*)t;
    float mn = 3.402823466e38f, mx = -3.402823466e38f;
    const int n4 = (IMG_H * IMG_W) / 4;
    for (int i = gid; i < n4; i += 64 * 256) {
        float4 v = t4[i];
        mn = fminf(mn, fminf(fminf(v.x, v.y), fminf(v.z, v.w)));
        mx = fmaxf(mx, fmaxf(fmaxf(v.x, v.y), fmaxf(v.z, v.w)));
    }
    smin[tid] = mn; smax[tid] = mx;
    __syncthreads();
    for (int s = 128; s > 0; s >>= 1) {
        if (tid < s) {
            smin[tid] = fminf(smin[tid], smin[tid + s]);
            smax[tid] = fmaxf(smax[tid], smax[tid + s]);
        }
        __syncthreads();
    }
    if (tid == 0) {
        atomicMin(&wsu[0], __float_as_uint(smin[0]));
        atomicMax(&wsu[1], __float_as_uint(smax[0]));
    }
}

// ---------------------------------------------------------------------------
// Kernel 2: one block per (b, w-split).
//   out[b,h,w] = a[h]*env[w] + bias[h] + scale[h]*target[h,w]
// computed as K=2 rank update on V_WMMA_F32_16X16X4_F32:
//   K0: a[h]   x env[w]
//   K1: bias[h] x 1      (both candidate K=1 B slots set to 1; A K2/K3 = 0)
//   C : scale[h]*target[h,w]   (single v_mul per element)
// ---------------------------------------------------------------------------
__global__ __launch_bounds__(256) void synth_blend_kernel(const float* __restrict__ params,
                                                          const float* __restrict__ target,
                                                          const float* __restrict__ ws,
                                                          float* __restrict__ out) {
    __shared__ float s_a[IMG_H];      // blend[h] * freq_prof[h] / mx
    __shared__ float s_scale[IMG_H];  // (1-blend[h]) * inv_range
    __shared__ float s_bias[IMG_H];   // -tmin * scale[h]
    __shared__ float s_red[256];

    const int b   = blockIdx.x;
    const int tid = threadIdx.x;

    // ---- derived parameters ----
    const float* pp = params + b * 11;
    const float p0 = pp[0], p1 = pp[1], p2 = pp[2], p3 = pp[3];
    const float fund      = pp[4]  * 0.7f + 0.1f;
    const float spread    = pp[5]  * 0.7f + 0.3f;
    const float p_nh      = pp[6]  * 0.8f + 0.2f;
    const float decay     = pp[7]  * 0.6f + 0.3f;
    const float attack    = pp[8]  * 0.5f;
    const float sustain   = pp[9]  * 0.5f + 0.5f;
    const float intensity = pp[10] * 0.5f + 0.5f;

    const float tmin = ws[0];
    const float tmax = ws[1];
    const float inv_range = 1.0f / (tmax - tmin + 1e-8f);

    float fb = floorf(fund * (float)IMG_H);
    fb = fminf(fmaxf(fb, 0.0f), (float)(IMG_H - 1));
    const float nh      = floorf(2.0f + p_nh * 8.0f);
    const float inv_att = 1.0f / (attack + 0.01f);
    const float sus3    = fminf(fmaxf(sustain, 0.3f), 1.0f);

    // ---- freq_prof for h = tid ----
    const float hf = (float)tid;
    float fp = 0.0f;
#pragma unroll
    for (int k = 0; k < 10; ++k) {
        const float kf = (float)k;
        const float hb = floorf(fb + kf * spread * (float)IMG_H / nh);
        const bool valid = (kf < nh) && (hb < (float)IMG_H);
        const float amp = intensity / (1.0f + kf * decay);
        const float d = (hf - hb) * 0.5f;
        const float g = __expf(-0.5f * d * d);
        fp += valid ? amp * g : 0.0f;
    }

    // ---- block max of freq_prof ----
    s_red[tid] = fp;
    __syncthreads();
    for (int s = 128; s > 0; s >>= 1) {
        if (tid < s) s_red[tid] = fmaxf(s_red[tid], s_red[tid + s]);
        __syncthreads();
    }
    const float maxfp = s_red[0];

    // env monotonic non-decreasing in w -> global max at w = W-1
    const float env_max = fminf(((float)(IMG_W - 1) / (float)IMG_W) * inv_att, 1.0f) * sus3;
    const float mx = maxfp * env_max;
    const float inv_mx = (mx > 0.0f) ? (1.0f / mx) : 1.0f;

    // ---- blend[h] for h = tid ----
    const float fbin = (float)tid * (1.0f / (float)(IMG_H - 1));
    const float sel = (fbin < 0.33f) ? p1 : ((fbin < 0.66f) ? p2 : p3);
    float blend = p0 + 0.3f * (sel - 0.5f);
    blend = fminf(fmaxf(blend, 0.0f), 1.0f);

    const float scale = (1.0f - blend) * inv_range;
    s_a[tid]     = blend * fp * inv_mx;
    s_scale[tid] = scale;
    s_bias[tid]  = -tmin * scale;
    __syncthreads();

    // ---- tile loop: each wave owns 2 h-tiles; inner loop over 16 w-tiles ----
    const int lane = tid & 31;
    const int wave = tid >> 5;
    const int n    = lane & 15;
    const int half = lane >> 4;
    const bool lo  = (lane < 16);

    float* __restrict__ outb = out + (size_t)b * IMG_H * IMG_W;
    const int wt0 = blockIdx.y * WT_PER_BLK;
    const float cw = inv_att * (1.0f / (float)IMG_W);

#pragma unroll
    for (int hh = 0; hh < 2; ++hh) {
        const int htile = wave + hh * 8;      // 0..15, disjoint across waves
        const int h0 = htile * 16;

        // A matrix (16x4): K0 col = a[h], K1 col = bias[h]; K2/K3 = 0
        v2f Am;
        Am.x = lo ? s_a[h0 + n]    : 0.0f;
        Am.y = lo ? s_bias[h0 + n] : 0.0f;

        const int mbase = h0 + half * 8;      // C/D rows for this lane half
        float sc[8];
#pragma unroll
        for (int r = 0; r < 8; ++r) sc[r] = s_scale[mbase + r];

        for (int wtile = wt0; wtile < wt0 + WT_PER_BLK; ++wtile) {
            const int wcol = wtile * 16 + n;

            // B matrix (4x16): K0 row = env; both candidate K1 slots = 1.0
            const float env = fminf((float)wcol * cw, 1.0f) * sus3;
            v2f Bm;
            Bm.x = lo ? env  : 1.0f;
            Bm.y = lo ? 1.0f : 0.0f;

            // C = scale[h] * target[h,w]  (one mul per element)
            v8f C;
#pragma unroll
            for (int r = 0; r < 8; ++r)
                C[r] = target[(size_t)(mbase + r) * IMG_W + wcol] * sc[r];

            v8f D = __builtin_amdgcn_wmma_f32_16x16x4_f32(
                /*neg_a=*/false, Am, /*neg_b=*/false, Bm,
                /*c_mod=*/(short)0, C, /*reuse_a=*/false, /*reuse_b=*/false);

#pragma unroll
            for (int r = 0; r < 8; ++r)
                outb[(size_t)(mbase + r) * IMG_W + wcol] = D[r];
        }
    }
}

// ---------------------------------------------------------------------------
extern "C" void kernel_launch(void* const* d_in, const int* in_sizes, int n_in,
                              void* d_out, int out_size, void* d_ws, size_t ws_size,
                              hipStream_t stream) {
    const float* params = (const float*)d_in[0];   // (256, 11) f32
    const float* target = (const float*)d_in[1];   // (256, 1024) f32
    float* out = (float*)d_out;                    // (256, 256, 1024) f32
    float* ws  = (float*)d_ws;
    unsigned int* wsu = (unsigned int*)d_ws;

    init_ws_kernel<<<1, 1, 0, stream>>>(wsu);
    minmax_kernel<<<64, 256, 0, stream>>>(target, wsu);
    synth_blend_kernel<<<dim3(NB, WSPLIT), 256, 0, stream>>>(params, target, ws, out);
}